// TransA_26027501814280
// MI455X (gfx1250) — compile-verified
//
#include <hip/hip_runtime.h>

#define BS   4096
#define SEQ  8192
#define HID  128
#define RELS 1000

typedef __attribute__((ext_vector_type(2))) float v2f;
typedef __attribute__((ext_vector_type(4))) float v4f;
typedef __attribute__((ext_vector_type(8))) float v8f;

__device__ __forceinline__ float waveReduce(float v) {
  #pragma unroll
  for (int off = 16; off > 0; off >>= 1)
    v += __shfl_down(v, off, 32);
  return v;
}

// ---------------- Kernel 1: gather + error + norm partials ----------------
// one block (128 threads) per sequence row i in [0, 8192).
// err is written in fused-A layout: errAB[j][0:128]=ep_j, errAB[j][128:256]=en_j
__global__ void k_gather_err(const int* __restrict__ inp,
                             const float* __restrict__ ent,
                             const float* __restrict__ rel,
                             float* __restrict__ errAB,
                             float* __restrict__ p_h2,
                             float* __restrict__ p_r2,
                             float* __restrict__ p_t2) {
  int i = blockIdx.x;
  int t = threadIdx.x;            // 0..127 = hidden index
  int hi = inp[i * 3 + 0];
  int ri = inp[i * 3 + 1];
  int ti = inp[i * 3 + 2];
  float hv = ent[hi * HID + t];
  float rv = rel[ri * HID + t];
  float tv = ent[ti * HID + t];
  int jj  = i & (BS - 1);         // pair row
  int off = (i >> 12) << 7;       // 0 for ep-half, 128 for en-half
  errAB[jj * 256 + off + t] = fabsf(hv + rv - tv);

  __shared__ float sh[12];
  float a = waveReduce(hv * hv);
  float b = waveReduce(rv * rv);
  float c = waveReduce(tv * tv);
  int lane = t & 31, w = t >> 5;
  if (lane == 0) { sh[w] = a; sh[4 + w] = b; sh[8 + w] = c; }
  __syncthreads();
  if (t == 0) {
    p_h2[i] = sh[0] + sh[1] + sh[2] + sh[3];
    p_r2[i] = sh[4] + sh[5] + sh[6] + sh[7];
    p_t2[i] = sh[8] + sh[9] + sh[10] + sh[11];
  }
}

// ---------------- Kernel 2: per-pair dots, fused B matrix, wr, winner -----
// one block (128 threads) per j in [0, 4096).
// vpn[j][0:128] = vp_j = relWr_j@ep_j ; vpn[j][128:256] = -vn_j
__global__ void k_pair(const float* __restrict__ errAB,
                       const int* __restrict__ inp,
                       float* __restrict__ vpn,
                       float* __restrict__ p_wr,
                       int* __restrict__ winner) {
  int j = blockIdx.x;
  int t = threadIdx.x;
  float ep = errAB[j * 256 + t];
  float en = errAB[j * 256 + 128 + t];

  __shared__ float sh[12];
  __shared__ float sred[3];
  float a = waveReduce(ep * ep);
  float b = waveReduce(en * en);
  float c = waveReduce(ep * en);
  int lane = t & 31, w = t >> 5;
  if (lane == 0) { sh[w] = a; sh[4 + w] = b; sh[8 + w] = c; }
  __syncthreads();
  if (t == 0) {
    float spp = sh[0] + sh[1] + sh[2] + sh[3];
    float snn = sh[4] + sh[5] + sh[6] + sh[7];
    float spn = sh[8] + sh[9] + sh[10] + sh[11];
    sred[0] = spp; sred[1] = snn; sred[2] = spn;
    p_wr[j] = snn * snn + spp * spp - 2.0f * spn * spn;   // ||relWr_j||_F^2
    atomicMax(&winner[inp[j * 3 + 1]], j);                 // last-wins scatter
  }
  __syncthreads();
  float spp = sred[0], snn = sred[1], spn = sred[2];
  vpn[j * 256 + t]       = en * spn - ep * spp;   //  vp_j
  vpn[j * 256 + 128 + t] = ep * spn - en * snn;   // -vn_j (negation folded in)
}

// ---------------- Kernel: init winner table ----------------
__global__ void k_init(int* __restrict__ winner) {
  int i = blockIdx.x * blockDim.x + threadIdx.x;
  if (i < RELS) winner[i] = -1;
}

// ---------------- Kernel 3: fused GEMM + relu-sum (WMMA f32) --------------
// S = A @ B^T(src row-major), A = errAB (M=4096,K=256), Bsrc = vpn (N=4096,K=256)
// block = 256 threads (8 waves), 128x128 output tile.
// A fragments: direct global->VGPR (L2-resident). B: LDS, pair-interleaved v2f.
__global__ __launch_bounds__(256) void k_gemm_margin(
    const float* __restrict__ A,
    const float* __restrict__ Bsrc,
    float* __restrict__ p_margin) {
  constexpr int BPAD = 144;          // v2f row stride; 2*144 mod 64 == 32 banks
  __shared__ v2f  Bp[16 * BPAD];     // 16 k-pairs x 128 n (padded) = 18KB
  __shared__ float wred[8];

  int tid = threadIdx.x, lane = tid & 31, wave = tid >> 5;
  int i0 = blockIdx.y * 128, j0 = blockIdx.x * 128;

  // WMMA lane mapping (ISA 16x16x4 f32 layout)
  int m  = lane & 15;                // matrix row (A) / col (B)
  int kq = (lane >> 4) << 1;         // K sub-pair: 0 or 2
  int kh = lane >> 4;                // k-pair parity select
  const float* arow = A + (i0 + wave * 16 + m) * 256;

  // B staging mapping: thread -> n = tid&127 ; kp range = (tid>>7)*8 .. +7
  int bn = tid & 127;
  int bg = (tid >> 7) * 8;
  const float* brow = Bsrc + (j0 + bn) * 256 + bg * 2;

  v8f zero = {0.f, 0.f, 0.f, 0.f, 0.f, 0.f, 0.f, 0.f};
  v8f acc[8];
  #pragma unroll
  for (int n = 0; n < 8; n++) acc[n] = zero;

  for (int kc = 0; kc < 256; kc += 32) {
    // ---- A fragment: 8 aligned b64 loads per lane, issued first ----
    v2f afrag[8];
    #pragma unroll
    for (int ks = 0; ks < 8; ks++)
      afrag[ks] = *(const v2f*)(arow + kc + ks * 4 + kq);

    // ---- stage B tile: 4 x global b128 + 8 x ds_store_b64 per thread ----
    #pragma unroll
    for (int u = 0; u < 4; u++) {
      v4f q = *(const v4f*)(brow + kc + 4 * u);          // kp = bg+2u, bg+2u+1
      v2f lo; lo.x = q.x; lo.y = q.y;
      v2f hi; hi.x = q.z; hi.y = q.w;
      Bp[(bg + 2 * u)     * BPAD + bn] = lo;
      Bp[(bg + 2 * u + 1) * BPAD + bn] = hi;
    }
    __syncthreads();

    // ---- 64 WMMAs per wave per chunk; B operand = single ds_load_b64 ----
    #pragma unroll
    for (int ks = 0; ks < 8; ks++) {
      int kpair = ks * 2 + kh;       // (ks*4 + kq) >> 1
      const v2f* brd = &Bp[kpair * BPAD + m];
      #pragma unroll
      for (int nt = 0; nt < 8; nt++) {
        v2f b = brd[nt * 16];
        acc[nt] = __builtin_amdgcn_wmma_f32_16x16x4_f32(
            false, afrag[ks], false, b, (short)0, acc[nt], false, false);
      }
    }
    __syncthreads();
  }

  // epilogue: sum relu(S + MARGIN), fixed-order deterministic reduction
  float local = 0.f;
  #pragma unroll
  for (int nt = 0; nt < 8; nt++)
    #pragma unroll
    for (int e = 0; e < 8; e++) {
      float s = acc[nt][e] + 1.0f;      // MARGIN
      local += (s > 0.f) ? s : 0.f;
    }
  local = waveReduce(local);
  if (lane == 0) wred[wave] = local;
  __syncthreads();
  if (tid == 0) {
    float tot = 0.f;
    #pragma unroll
    for (int w = 0; w < 8; w++) tot += wred[w];
    p_margin[blockIdx.y * gridDim.x + blockIdx.x] = tot;
  }
}

// ---------------- Kernel 4: scatter winning relWr rows into Wr_new --------
__global__ void k_scatter(const float* __restrict__ errAB,
                          const int* __restrict__ winner,
                          float* __restrict__ Wr_out) {
  int rel = blockIdx.x;
  int j = winner[rel];
  if (j < 0) return;                 // untouched rows keep copied Wr (zeros)
  __shared__ float sepn[256];        // [0:128]=ep_j, [128:256]=en_j
  int t = threadIdx.x;               // 256
  sepn[t] = errAB[j * 256 + t];
  __syncthreads();
  const float* sep = sepn;
  const float* sen = sepn + 128;
  float* dst = Wr_out + (long)rel * HID * HID;
  #pragma unroll
  for (int it = 0; it < 64; it++) {
    int idx = t + it * 256;
    int hh = idx >> 7, kk = idx & 127;
    dst[idx] = sen[hh] * sen[kk] - sep[hh] * sep[kk];
  }
}

// ---------------- Kernel 5: final deterministic combine -------------------
__device__ float blockSum256(float v, float* sh) {
  int t = threadIdx.x;
  sh[t] = v; __syncthreads();
  for (int s = 128; s > 0; s >>= 1) {
    if (t < s) sh[t] += sh[t + s];
    __syncthreads();
  }
  float r = sh[0];
  __syncthreads();
  return r;
}

__global__ void k_final(const float* __restrict__ p_h2,
                        const float* __restrict__ p_r2,
                        const float* __restrict__ p_t2,
                        const float* __restrict__ p_wr,
                        const float* __restrict__ p_margin,
                        float* __restrict__ out) {
  __shared__ float sh[256];
  int t = threadIdx.x;
  float h2 = 0.f, r2 = 0.f, t2 = 0.f, wr = 0.f, mg = 0.f;
  for (int i = t; i < SEQ; i += 256) { h2 += p_h2[i]; r2 += p_r2[i]; t2 += p_t2[i]; }
  for (int i = t; i < BS;  i += 256) wr += p_wr[i];
  for (int i = t; i < 1024; i += 256) mg += p_margin[i];
  h2 = blockSum256(h2, sh);
  r2 = blockSum256(r2, sh);
  t2 = blockSum256(t2, sh);
  wr = blockSum256(wr, sh);
  mg = blockSum256(mg, sh);
  if (t == 0) {
    out[0] = mg / (float)BS
           + 0.01f * sqrtf(wr)
           + 0.01f * (sqrtf(h2) + sqrtf(r2) + sqrtf(t2));
  }
}

// ---------------- launcher ----------------
extern "C" void kernel_launch(void* const* d_in, const int* in_sizes, int n_in,
                              void* d_out, int out_size, void* d_ws, size_t ws_size,
                              hipStream_t stream) {
  const int*   inp = (const int*)d_in[0];
  const float* ent = (const float*)d_in[1];
  const float* rel = (const float*)d_in[2];
  const float* Wr  = (const float*)d_in[3];

  float* out    = (float*)d_out;
  float* Wr_out = out + 1;                       // (loss, Wr_new) concat

  float* ws       = (float*)d_ws;
  float* errAB    = ws;                          // 4096*256
  float* vpn      = errAB + BS * 256;            // 4096*256
  float* p_h2     = vpn + BS * 256;              // 8192
  float* p_r2     = p_h2 + SEQ;
  float* p_t2     = p_r2 + SEQ;
  float* p_wr     = p_t2 + SEQ;                  // 4096
  float* p_margin = p_wr + BS;                   // 1024
  int*   winner   = (int*)(p_margin + 1024);     // 1000

  // Wr_new starts as a copy of Wr (zeros); winners overwrite below.
  hipMemcpyAsync(Wr_out, Wr, (size_t)RELS * HID * HID * sizeof(float),
                 hipMemcpyDeviceToDevice, stream);

  k_init<<<(RELS + 255) / 256, 256, 0, stream>>>(winner);
  k_gather_err<<<SEQ, 128, 0, stream>>>(inp, ent, rel, errAB, p_h2, p_r2, p_t2);
  k_pair<<<BS, 128, 0, stream>>>(errAB, inp, vpn, p_wr, winner);

  dim3 g(32, 32);
  k_gemm_margin<<<g, 256, 0, stream>>>(errAB, vpn, p_margin);

  k_scatter<<<RELS, 256, 0, stream>>>(errAB, winner, Wr_out);
  k_final<<<1, 256, 0, stream>>>(p_h2, p_r2, p_t2, p_wr, p_margin, out);
}